// CrossNet_31997506355235
// MI455X (gfx1250) — compile-verified
//
#include <hip/hip_runtime.h>
#include <hip/hip_bf16.h>

typedef __attribute__((ext_vector_type(2))) float v2f;
typedef __attribute__((ext_vector_type(4))) float f4;
typedef __attribute__((ext_vector_type(8))) float v8f;

#define NCOLS 4096
#define NROWS 4096
#define TILE_R 16
#define NTHREADS 256
#define CHUNKS 64  // NCOLS / (16 col-groups * 4 floats)

// One WG per 16-row tile. Thread (r = tid>>4, g = tid&15) streams row (tile*16+r),
// columns {ch*64 + g*4 .. +3} for ch in 0..63.
// Phase 1: per-thread partials of d_j = x_row . w_j  (and, for r==0, c_jk = b_j . w_k)
//          into LDS M[64][16] (sets: d0,d1,d2,c).
// Phase 1b: wave0 reduces M row-sums with chained V_WMMA_F32_16X16X4_F32 (B = ones),
//           then solves the scalar recurrence -> per-row scales T0,T1,T2.
// Phase 2: out_k[row,col] = x*T_k + prefix_b_k[col], written with non-temporal b128.
__global__ __launch_bounds__(NTHREADS)
void crossnet_fused(const float* __restrict__ x,
                    const float* __restrict__ w,
                    const float* __restrict__ bb,
                    float* __restrict__ out) {
  __shared__ float M[64][16];   // partial sums: rows [0..15]=d0, [16..31]=d1, [32..47]=d2, [48..50]=c01,c02,c12
  __shared__ float R[64];       // WMMA-reduced row sums
  __shared__ float T[TILE_R][4];

  const int tid = threadIdx.x;
  const int g = tid & 15;
  const int r = tid >> 4;
  const int row = blockIdx.x * TILE_R + r;
  const float* __restrict__ xr = x + (size_t)row * NCOLS;

  // zero M (256 threads x 4 floats = 1024 floats)
  {
    f4 z = {0.f, 0.f, 0.f, 0.f};
    ((f4*)&M[0][0])[tid] = z;
  }
  __syncthreads();

  // ---- Phase 1: streaming partial dot products ----
  float d0 = 0.f, d1 = 0.f, d2 = 0.f;
  float c01 = 0.f, c02 = 0.f, c12 = 0.f;
#pragma unroll 4
  for (int ch = 0; ch < CHUNKS; ++ch) {
    const int col = ch * 64 + g * 4;
    __builtin_prefetch(xr + col + 1024, 0, 0);  // global_prefetch_b8, ~4KB ahead
    f4 xv = *(const f4*)(xr + col);
    f4 w0 = *(const f4*)(w + col);
    f4 w1 = *(const f4*)(w + NCOLS + col);
    f4 w2 = *(const f4*)(w + 2 * NCOLS + col);
    d0 += xv.x * w0.x + xv.y * w0.y + xv.z * w0.z + xv.w * w0.w;
    d1 += xv.x * w1.x + xv.y * w1.y + xv.z * w1.z + xv.w * w1.w;
    d2 += xv.x * w2.x + xv.y * w2.y + xv.z * w2.z + xv.w * w2.w;
    if (r == 0) {  // c_jk are row-independent; only one row-strip computes them
      f4 b0 = *(const f4*)(bb + col);
      f4 b1 = *(const f4*)(bb + NCOLS + col);
      c01 += b0.x * w1.x + b0.y * w1.y + b0.z * w1.z + b0.w * w1.w;
      c02 += b0.x * w2.x + b0.y * w2.y + b0.z * w2.z + b0.w * w2.w;
      c12 += b1.x * w2.x + b1.y * w2.y + b1.z * w2.z + b1.w * w2.w;
    }
  }
  M[r][g] = d0;
  M[16 + r][g] = d1;
  M[32 + r][g] = d2;
  if (r == 0) { M[48][g] = c01; M[49][g] = c02; M[50][g] = c12; }
  __syncthreads();

  // ---- Phase 1b: wave0 reduces row sums of M via WMMA f32 16x16x4, B = ones ----
  if (tid < 32) {
    const int m = tid & 15;   // A-matrix row (M dimension)
    const int h = tid >> 4;   // lane half selects K pair {2h, 2h+1}
    v2f ones; ones[0] = 1.f; ones[1] = 1.f;  // B all-ones: layout-independent
#pragma unroll
    for (int s = 0; s < 4; ++s) {
      v8f acc = {0.f, 0.f, 0.f, 0.f, 0.f, 0.f, 0.f, 0.f};
#pragma unroll
      for (int q = 0; q < 4; ++q) {
        // 32-bit A 16x4 layout: lanes 0-15 -> K=0 (v0), K=1 (v1); lanes 16-31 -> K=2,3
        const int gg = q * 4 + 2 * h;
        v2f a;
        a[0] = M[s * 16 + m][gg];
        a[1] = M[s * 16 + m][gg + 1];
        acc = __builtin_amdgcn_wmma_f32_16x16x4_f32(
            false, a, false, ones, (short)0, acc, false, false);
      }
      // D[m][n] = row-sum(M[s*16+m]) replicated over n. Extract column n=0:
      // lane 0 holds m=0..7 in acc[0..7]; lane 16 holds m=8..15.
      if (m == 0) {
        const int base = s * 16 + h * 8;
#pragma unroll
        for (int i = 0; i < 8; ++i) R[base + i] = acc[i];
      }
    }
    // scalar recurrence:  s_k = d_k*(1+Σ s_j) + Σ c_jk ;  T_k = 1 + Σ_{j<=k} s_j
    if (tid < 16) {
      const float P0 = R[tid], P1 = R[16 + tid], P2 = R[32 + tid];
      const float k01 = R[48], k02 = R[49], k12 = R[50];
      const float s0 = P0;
      const float t0 = 1.f + s0;
      const float s1 = P1 * t0 + k01;
      const float t1 = t0 + s1;
      const float s2 = P2 * t1 + (k02 + k12);
      const float t2 = t1 + s2;
      T[tid][0] = t0; T[tid][1] = t1; T[tid][2] = t2;
    }
  }
  __syncthreads();

  // ---- Phase 2: stream outputs (x re-read hits L2; stores non-temporal) ----
  const float t0 = T[r][0], t1 = T[r][1], t2 = T[r][2];
  const f4 t0v = {t0, t0, t0, t0};
  const f4 t1v = {t1, t1, t1, t1};
  const f4 t2v = {t2, t2, t2, t2};
  const size_t PLANE = (size_t)NROWS * NCOLS;
  float* __restrict__ o0p = out + (size_t)row * NCOLS;
  float* __restrict__ o1p = o0p + PLANE;
  float* __restrict__ o2p = o1p + PLANE;
#pragma unroll 4
  for (int ch = 0; ch < CHUNKS; ++ch) {
    const int col = ch * 64 + g * 4;
    f4 xv = *(const f4*)(xr + col);
    f4 b0 = *(const f4*)(bb + col);
    f4 b1 = *(const f4*)(bb + NCOLS + col);
    f4 b2 = *(const f4*)(bb + 2 * NCOLS + col);
    f4 s01 = b0 + b1;
    f4 o0 = xv * t0v + b0;
    f4 o1 = xv * t1v + s01;
    f4 o2 = xv * t2v + (s01 + b2);
    __builtin_nontemporal_store(o0, (f4*)(o0p + col));
    __builtin_nontemporal_store(o1, (f4*)(o1p + col));
    __builtin_nontemporal_store(o2, (f4*)(o2p + col));
  }
}

extern "C" void kernel_launch(void* const* d_in, const int* in_sizes, int n_in,
                              void* d_out, int out_size, void* d_ws, size_t ws_size,
                              hipStream_t stream) {
  (void)in_sizes; (void)n_in; (void)out_size; (void)d_ws; (void)ws_size;
  const float* x = (const float*)d_in[0];   // [4096, 4096]
  const float* w = (const float*)d_in[1];   // [3, 4096, 1]
  const float* b = (const float*)d_in[2];   // [3, 4096, 1]
  float* out = (float*)d_out;               // 3 x [4096, 4096] concatenated
  dim3 grid(NROWS / TILE_R);
  dim3 block(NTHREADS);
  hipLaunchKernelGGL(crossnet_fused, grid, block, 0, stream, x, w, b, out);
}